// SparseUNet2D_25598005084682
// MI455X (gfx1250) — compile-verified
//
#include <hip/hip_runtime.h>
#include <math.h>

// ---------------------------------------------------------------------------
// Sparse UNet 2D for gfx1250 (MI455X). Convolutions run as implicit GEMM on
// V_WMMA_F32_16X16X4_F32 (f32 matrix pipe, exact precision vs reference).
// K dimension is permuted tap-major (k' = tap*Cin + c) so each K-step of 4
// shares one tap: boundary handling hoisted out of the channel loop, loads
// unconditional (clamped address * 0/1 mask), weights pre-transposed so the
// A fragment is one contiguous b64 load.
// Norm reductions are deterministic two-stage tree reductions (no fp atomics).
// ---------------------------------------------------------------------------

typedef float v2f __attribute__((ext_vector_type(2)));
typedef float v8f __attribute__((ext_vector_type(8)));

#define WPB 8            // waves per block in the conv kernel (256 threads)
#define BPC 64           // reduction blocks per channel (fixed -> deterministic)

// weight relayout: w[o][c][tap] -> wT[o][tap][c]
__global__ __launch_bounds__(256) void k_wtrans(
    const float* __restrict__ w, float* __restrict__ wT, int C, int O)
{
    int i = blockIdx.x * 256 + threadIdx.x;
    int T = O * C * 9;
    if (i >= T) return;
    int o   = i / (C * 9);
    int r   = i - o * C * 9;
    int c   = r / 9;
    int tap = r - c * 9;
    wT[((size_t)o * 9 + tap) * C + c] = w[i];
}

// ---------------------------------------------------------------------------
// conv3x3 (SAME) as implicit GEMM via WMMA f32 16x16x4, K order tap-major.
// One wave computes D[16 outch x 16 pixels] for one row y, 16 consecutive x.
// A fragment: lane l holds wT[oc=ocb*16+(l&15)][tap][c + 2*(l>>4) + {0,1}]
// B fragment: lane l holds x[c + 2*(l>>4) + {0,1}][yy][xx], px = xb + (l&15)
// D: reg r, lane l -> outch ocb*16 + r + 8*(l>>4), pixel xb + (l&15).
// Output stored pre-masked (inactive sites -> 0).
// ---------------------------------------------------------------------------
template <int CIN>
__global__ __launch_bounds__(256) void k_conv3x3_wmma(
    const float* __restrict__ x, const float* __restrict__ wT,
    const int* __restrict__ mask, float* __restrict__ out,
    int Cout, int H, int W, int totalTiles)
{
    int wave = blockIdx.x * WPB + (threadIdx.x >> 5);
    if (wave >= totalTiles) return;               // wave-uniform exit
    int lane = threadIdx.x & 31;
    int lx = lane & 15;
    int ly = lane >> 4;
    int nXT = W >> 4;
    int rowTiles = H * nXT;
    int ocb = wave / rowTiles;
    int rem = wave - ocb * rowTiles;
    int y   = rem / nXT;
    int xb  = (rem - y * nXT) << 4;
    const int K = CIN * 9;
    size_t HW = (size_t)H * W;

    const float* wrow = wT + (size_t)(ocb * 16 + lx) * K;  // A row (tap-major)
    const int c2 = ly << 1;                                // lane's K-pair offset
    v8f acc = {};

#pragma unroll
    for (int tap = 0; tap < 9; ++tap) {
        const int ty = tap / 3 - 1;
        const int tx = tap % 3 - 1;
        int yy = y + ty;
        int xx = xb + lx + tx;
        float inb = ((unsigned)yy < (unsigned)H && (unsigned)xx < (unsigned)W)
                        ? 1.0f : 0.0f;
        int yyc = min(max(yy, 0), H - 1);
        int xxc = min(max(xx, 0), W - 1);
        const float* bp = x + (size_t)yyc * W + xxc + (size_t)c2 * HW;
        const float* ap = wrow + tap * CIN + c2;
#pragma unroll
        for (int c = 0; c < CIN; c += 4) {
            v2f a, b;
            a.x = ap[0];
            a.y = ap[1];
            b.x = bp[0]  * inb;
            b.y = bp[HW] * inb;
            acc = __builtin_amdgcn_wmma_f32_16x16x4_f32(
                false, a, false, b, (short)0, acc, false, false);
            ap += 4;
            bp += (size_t)4 * HW;
        }
    }

    int px = xb + lx;
    float mm = mask[(size_t)y * W + px] ? 1.0f : 0.0f;
#pragma unroll
    for (int r = 0; r < 8; ++r) {
        int oc = ocb * 16 + r + (ly << 3);
        out[(size_t)oc * HW + (size_t)y * W + px] = acc[r] * mm;
    }
}

// ---------------------------------------------------------------------------
// Deterministic per-channel sum / sum-of-squares reduction (stage 1).
// grid = (BPC, C). Input buffer is pre-masked (zeros at inactive sites).
// ---------------------------------------------------------------------------
__global__ __launch_bounds__(256) void k_reduce_partial(
    const float* __restrict__ x, int S, float* __restrict__ part)
{
    int c = blockIdx.y;
    const float* xc = x + (size_t)c * S;
    float s = 0.0f, s2 = 0.0f;
    for (int i = blockIdx.x * 256 + threadIdx.x; i < S; i += BPC * 256) {
        float v = xc[i];
        s  += v;
        s2 += v * v;
    }
    __shared__ float sh[256], sh2[256];
    sh[threadIdx.x] = s; sh2[threadIdx.x] = s2;
    __syncthreads();
    for (int o = 128; o > 0; o >>= 1) {
        if (threadIdx.x < o) {
            sh[threadIdx.x]  += sh[threadIdx.x + o];
            sh2[threadIdx.x] += sh2[threadIdx.x + o];
        }
        __syncthreads();
    }
    if (threadIdx.x == 0) {
        size_t idx = ((size_t)c * BPC + blockIdx.x) * 2;
        part[idx]     = sh[0];
        part[idx + 1] = sh2[0];
    }
}

// stage 2: mean + inv-std per channel (fixed-order accumulation).
__global__ void k_stats(const float* __restrict__ part, int C,
                        const float* __restrict__ cnt, float eps,
                        float* __restrict__ stats)
{
    int c = threadIdx.x;
    if (c >= C) return;
    float s = 0.0f, s2 = 0.0f;
    for (int b = 0; b < BPC; ++b) {
        s  += part[((size_t)c * BPC + b) * 2];
        s2 += part[((size_t)c * BPC + b) * 2 + 1];
    }
    float n    = fmaxf(cnt[0], 1.0f);
    float mean = s / n;
    float var  = fmaxf(s2 / n - mean * mean, 0.0f);
    stats[c * 2]     = mean;
    stats[c * 2 + 1] = rsqrtf(var + eps);
}

// BatchNorm affine + leaky(slope) + mask
__global__ __launch_bounds__(256) void k_bn_apply(
    const float* __restrict__ h, const float* __restrict__ stats,
    const float* __restrict__ gw, const float* __restrict__ gb,
    const int* __restrict__ mask, float* __restrict__ out,
    int S, int C, float slope)
{
    long i = (long)blockIdx.x * 256 + threadIdx.x;
    if (i >= (long)C * S) return;
    int c = (int)(i / S);
    int p = (int)(i - (long)c * S);
    float v = (h[i] - stats[c * 2]) * stats[c * 2 + 1] * gw[c] + gb[c];
    v = v >= 0.0f ? v : slope * v;
    out[i] = mask[p] ? v : 0.0f;
}

// InstanceNorm (no affine) + residual add + leaky(0.01) + mask
__global__ __launch_bounds__(256) void k_in_res_apply(
    const float* __restrict__ h, const float* __restrict__ stats,
    const float* __restrict__ res, const int* __restrict__ mask,
    float* __restrict__ out, int S, int C)
{
    long i = (long)blockIdx.x * 256 + threadIdx.x;
    if (i >= (long)C * S) return;
    int c = (int)(i / S);
    int p = (int)(i - (long)c * S);
    float v = (h[i] - stats[c * 2]) * stats[c * 2 + 1] + res[i];
    v = v >= 0.0f ? v : 0.01f * v;
    out[i] = mask[p] ? v : 0.0f;
}

// sparse 2x2 maxpool (active sites only), output masked
__global__ __launch_bounds__(256) void k_maxpool(
    const float* __restrict__ x, const int* __restrict__ mi,
    const int* __restrict__ mo, float* __restrict__ out,
    int C, int Ho, int Wo)
{
    long i = (long)blockIdx.x * 256 + threadIdx.x;
    long T = (long)C * Ho * Wo;
    if (i >= T) return;
    int c = (int)(i / ((long)Ho * Wo));
    int p = (int)(i - (long)c * Ho * Wo);
    int y = p / Wo, xx = p - y * Wo;
    if (!mo[p]) { out[i] = 0.0f; return; }
    int Hi = Ho * 2, Wi = Wo * 2;
    float best = -1e30f;
#pragma unroll
    for (int dy = 0; dy < 2; ++dy)
#pragma unroll
        for (int dx = 0; dx < 2; ++dx) {
            int yy = 2 * y + dy, xi = 2 * xx + dx;
            size_t idx = (size_t)yy * Wi + xi;
            float v = mi[idx] ? x[(size_t)c * Hi * Wi + idx] : -1e30f;
            best = fmaxf(best, v);
        }
    out[i] = best;
}

// mask 2x2 "any" pool
__global__ __launch_bounds__(256) void k_mask_pool(
    const int* __restrict__ mi, int* __restrict__ mo, int Ho, int Wo)
{
    int p = blockIdx.x * 256 + threadIdx.x;
    if (p >= Ho * Wo) return;
    int y = p / Wo, xx = p - y * Wo;
    int Wi = Wo * 2;
    int a = mi[(size_t)(2 * y) * Wi + 2 * xx] | mi[(size_t)(2 * y) * Wi + 2 * xx + 1]
          | mi[(size_t)(2 * y + 1) * Wi + 2 * xx] | mi[(size_t)(2 * y + 1) * Wi + 2 * xx + 1];
    mo[p] = a ? 1 : 0;
}

// mask nearest 2x upsample (coordinate subdivision)
__global__ __launch_bounds__(256) void k_mask_up(
    const int* __restrict__ mi, int* __restrict__ mo, int H, int W)
{
    int p = blockIdx.x * 256 + threadIdx.x;
    if (p >= H * W) return;
    int y = p / W, xx = p - y * W;
    mo[p] = mi[(size_t)(y >> 1) * (W >> 1) + (xx >> 1)];
}

// count active sites in a mask (single block, deterministic)
__global__ __launch_bounds__(1024) void k_count(
    const int* __restrict__ m, int S, float* __restrict__ out)
{
    __shared__ float sh[1024];
    float s = 0.0f;
    for (int i = threadIdx.x; i < S; i += 1024) s += m[i] ? 1.0f : 0.0f;
    sh[threadIdx.x] = s;
    __syncthreads();
    for (int o = 512; o > 0; o >>= 1) {
        if (threadIdx.x < o) sh[threadIdx.x] += sh[threadIdx.x + o];
        __syncthreads();
    }
    if (threadIdx.x == 0) out[0] = sh[0];
}

// decoder mask counts: cm0 = 4*n2, cm1 = 16*n2 (exact for 2x nearest upsample)
__global__ void k_derive(float* __restrict__ cnt)
{
    if (threadIdx.x == 0) { cnt[3] = 4.0f * cnt[2]; cnt[4] = 16.0f * cnt[2]; }
}

// xin0 = mask ? features : 0
__global__ __launch_bounds__(256) void k_mask_features(
    const float* __restrict__ f, const int* __restrict__ m,
    float* __restrict__ out, int S, int C)
{
    long i = (long)blockIdx.x * 256 + threadIdx.x;
    if (i >= (long)C * S) return;
    int p = (int)(i % S);
    out[i] = m[p] ? f[i] : 0.0f;
}

// ConvTranspose2d(k=2,s=2) + skip add + mask  (H,W = OUTPUT dims)
__global__ __launch_bounds__(256) void k_upsample_add(
    const float* __restrict__ x, const float* __restrict__ wup,
    const float* __restrict__ skip, const int* __restrict__ cm,
    float* __restrict__ out, int Cin, int Cout, int H, int W)
{
    long i = (long)blockIdx.x * 256 + threadIdx.x;
    long T = (long)Cout * H * W;
    if (i >= T) return;
    int o = (int)(i / ((long)H * W));
    int p = (int)(i - (long)o * H * W);
    if (!cm[p]) { out[i] = 0.0f; return; }
    int y = p / W, xx = p - y * W;
    int dy = y & 1, dx = xx & 1;
    int ys = y >> 1, xs = xx >> 1;
    int Hs = H >> 1, Ws = W >> 1;
    float acc = skip[i];
    for (int c = 0; c < Cin; ++c)
        acc += x[(size_t)c * Hs * Ws + (size_t)ys * Ws + xs] *
               wup[(((size_t)c * Cout + o) * 2 + dy) * 2 + dx];
    out[i] = acc;
}

// predict head, pass 1: partial sums of h = w1 @ x (recomputed on the fly)
__global__ __launch_bounds__(256) void k_pred_stats(
    const float* __restrict__ x, const float* __restrict__ w1,
    int Cin, int S, float* __restrict__ part)
{
    int o = blockIdx.y;
    float s = 0.0f, s2 = 0.0f;
    for (int p = blockIdx.x * 256 + threadIdx.x; p < S; p += BPC * 256) {
        float a = 0.0f;
        for (int c = 0; c < Cin; ++c)
            a += w1[(size_t)o * Cin + c] * x[(size_t)c * S + p];
        s += a; s2 += a * a;
    }
    __shared__ float sh[256], sh2[256];
    sh[threadIdx.x] = s; sh2[threadIdx.x] = s2;
    __syncthreads();
    for (int q = 128; q > 0; q >>= 1) {
        if (threadIdx.x < q) {
            sh[threadIdx.x]  += sh[threadIdx.x + q];
            sh2[threadIdx.x] += sh2[threadIdx.x + q];
        }
        __syncthreads();
    }
    if (threadIdx.x == 0) {
        size_t idx = ((size_t)o * BPC + blockIdx.x) * 2;
        part[idx] = sh[0]; part[idx + 1] = sh2[0];
    }
}

// predict head, pass 2: logits = w2 @ leaky(IN(w1 @ x)), background [1, 0]
__global__ __launch_bounds__(256) void k_pred_out(
    const float* __restrict__ x, const float* __restrict__ w1,
    const float* __restrict__ w2, const float* __restrict__ stats,
    const int* __restrict__ cm, float* __restrict__ out,
    int Cin, int O, int S)
{
    int p = blockIdx.x * 256 + threadIdx.x;
    if (p >= S) return;
    if (!cm[p]) { out[p] = 1.0f; out[(size_t)S + p] = 0.0f; return; }
    float l0 = 0.0f, l1 = 0.0f;
    for (int o = 0; o < O; ++o) {
        float a = 0.0f;
        for (int c = 0; c < Cin; ++c)
            a += w1[(size_t)o * Cin + c] * x[(size_t)c * S + p];
        float g = (a - stats[2 * o]) * stats[2 * o + 1];
        g = g >= 0.0f ? g : 0.01f * g;
        l0 += w2[o] * g;
        l1 += w2[O + o] * g;
    }
    out[p] = l0;
    out[(size_t)S + p] = l1;
}

// ---------------------------------------------------------------------------
// Host-side helpers
// ---------------------------------------------------------------------------
static void launch_conv(hipStream_t st, const float* x, const float* w, float* wT,
                        const int* mask, float* out, int Cin, int Cout, int H, int W)
{
    int T = Cout * Cin * 9;
    k_wtrans<<<(T + 255) / 256, 256, 0, st>>>(w, wT, Cin, Cout);
    int tiles = (Cout / 16) * H * (W / 16);
    int g = (tiles + WPB - 1) / WPB;
    if (Cin == 16)
        k_conv3x3_wmma<16><<<g, 256, 0, st>>>(x, wT, mask, out, Cout, H, W, tiles);
    else if (Cin == 32)
        k_conv3x3_wmma<32><<<g, 256, 0, st>>>(x, wT, mask, out, Cout, H, W, tiles);
    else
        k_conv3x3_wmma<64><<<g, 256, 0, st>>>(x, wT, mask, out, Cout, H, W, tiles);
}

static void res_block(hipStream_t st, const float* xin, const int* mask,
                      int Cin, int Cout, int H, int W,
                      const float* c1, const float* bn1w, const float* bn1b,
                      const float* c2, const float* bn2w, const float* bn2b,
                      const float* c3,
                      float* A, float* B, float* R, float* WT, float* out,
                      float* part, float* stats, const float* cnt)
{
    int S = H * W;
    long N = (long)Cout * S;
    int ebl = (int)((N + 255) / 256);

    // conv1 -> BN(eps 1e-4) + lrelu 0.333 -> R (residual)
    launch_conv(st, xin, c1, WT, mask, A, Cin, Cout, H, W);
    k_reduce_partial<<<dim3(BPC, Cout), 256, 0, st>>>(A, S, part);
    k_stats<<<1, 64, 0, st>>>(part, Cout, cnt, 1e-4f, stats);
    k_bn_apply<<<ebl, 256, 0, st>>>(A, stats, bn1w, bn1b, mask, R, S, Cout, 0.333f);
    // conv2 -> BN + lrelu 0.333 (in place in B)
    launch_conv(st, R, c2, WT, mask, B, Cout, Cout, H, W);
    k_reduce_partial<<<dim3(BPC, Cout), 256, 0, st>>>(B, S, part);
    k_stats<<<1, 64, 0, st>>>(part, Cout, cnt, 1e-4f, stats);
    k_bn_apply<<<ebl, 256, 0, st>>>(B, stats, bn2w, bn2b, mask, B, S, Cout, 0.333f);
    // conv3 -> InstanceNorm(eps 1e-5) + residual + lrelu 0.01
    launch_conv(st, B, c3, WT, mask, A, Cout, Cout, H, W);
    k_reduce_partial<<<dim3(BPC, Cout), 256, 0, st>>>(A, S, part);
    k_stats<<<1, 64, 0, st>>>(part, Cout, cnt, 1e-5f, stats);
    k_in_res_apply<<<ebl, 256, 0, st>>>(A, stats, R, mask, out, S, Cout);
}

// d_in indices: JAX pytree order (dict keys sorted alphabetically)
enum {
    IN_FEAT = 0, IN_MASK = 1,
    DEC0_BN1B = 2, DEC0_BN1W, DEC0_BN2B, DEC0_BN2W, DEC0_C1, DEC0_C2, DEC0_C3, DEC0_UP,
    DEC1_BN1B = 10, DEC1_BN1W, DEC1_BN2B, DEC1_BN2W, DEC1_C1, DEC1_C2, DEC1_C3, DEC1_UP,
    ENC0_BN1B = 18, ENC0_BN1W, ENC0_BN2B, ENC0_BN2W, ENC0_C1, ENC0_C2, ENC0_C3,
    ENC1_BN1B = 25, ENC1_BN1W, ENC1_BN2B, ENC1_BN2W, ENC1_C1, ENC1_C2, ENC1_C3,
    ENC2_BN1B = 32, ENC2_BN1W, ENC2_BN2B, ENC2_BN2W, ENC2_C1, ENC2_C2, ENC2_C3,
    PRED0_W1 = 39, PRED0_W2, PRED1_W1 = 41, PRED1_W2
};

extern "C" void kernel_launch(void* const* d_in, const int* in_sizes, int n_in,
                              void* d_out, int out_size, void* d_ws, size_t ws_size,
                              hipStream_t stream)
{
    (void)in_sizes; (void)n_in; (void)out_size; (void)ws_size;
    const int S0 = 1024 * 1024, S1 = 512 * 512, S2 = 256 * 256;

    const float* feat = (const float*)d_in[IN_FEAT];
    const int*   M0   = (const int*)d_in[IN_MASK];
#define FP(i) ((const float*)d_in[i])

    // ---- workspace layout (floats) ----
    float* f = (float*)d_ws;
    size_t o = 0;
    float* X0 = f + o; o += (size_t)16 * S0;   // enc0 out (skip for dec1)
    float* X1 = f + o; o += (size_t)32 * S1;   // enc1 out (skip for dec0)
    float* X2 = f + o; o += (size_t)64 * S2;   // enc2 out (bottleneck)
    float* D0 = f + o; o += (size_t)32 * S1;   // dec0 out
    float* A  = f + o; o += (size_t)16 * S0;   // conv scratch 1/3
    float* B  = f + o; o += (size_t)16 * S0;   // xin / conv scratch 2
    float* R  = f + o; o += (size_t)16 * S0;   // residual
    float* WT = f + o; o += 40960;             // transposed conv weights
    int* M1  = (int*)(f + o); o += S1;         // masks
    int* M2  = (int*)(f + o); o += S2;
    int* CM0 = (int*)(f + o); o += S1;
    int* CM1 = (int*)(f + o); o += S0;
    float* CNT   = f + o; o += 8;              // n0,n1,n2,ncm0,ncm1
    float* PART  = f + o; o += (size_t)64 * BPC * 2;
    float* STATS = f + o; o += 128;

    float* xd_out  = (float*)d_out;            // [16,1024,1024]
    float* logitsF = xd_out + (size_t)16 * S0; // [2,1024,1024]
    float* logitsC = logitsF + (size_t)2 * S0; // [2,512,512]

    // ---- masks & active-site counts ----
    k_mask_pool<<<(S1 + 255) / 256, 256, 0, stream>>>(M0, M1, 512, 512);
    k_mask_pool<<<(S2 + 255) / 256, 256, 0, stream>>>(M1, M2, 256, 256);
    k_mask_up<<<(S1 + 255) / 256, 256, 0, stream>>>(M2, CM0, 512, 512);
    k_mask_up<<<(S0 + 255) / 256, 256, 0, stream>>>(CM0, CM1, 1024, 1024);
    k_count<<<1, 1024, 0, stream>>>(M0, S0, CNT + 0);
    k_count<<<1, 1024, 0, stream>>>(M1, S1, CNT + 1);
    k_count<<<1, 1024, 0, stream>>>(M2, S2, CNT + 2);
    k_derive<<<1, 32, 0, stream>>>(CNT);

    // ---- encoder ----
    k_mask_features<<<(int)(((long)16 * S0 + 255) / 256), 256, 0, stream>>>(feat, M0, B, S0, 16);
    res_block(stream, B, M0, 16, 16, 1024, 1024,
              FP(ENC0_C1), FP(ENC0_BN1W), FP(ENC0_BN1B),
              FP(ENC0_C2), FP(ENC0_BN2W), FP(ENC0_BN2B), FP(ENC0_C3),
              A, B, R, WT, X0, PART, STATS, CNT + 0);

    k_maxpool<<<(int)(((long)16 * S1 + 255) / 256), 256, 0, stream>>>(X0, M0, M1, B, 16, 512, 512);
    res_block(stream, B, M1, 16, 32, 512, 512,
              FP(ENC1_C1), FP(ENC1_BN1W), FP(ENC1_BN1B),
              FP(ENC1_C2), FP(ENC1_BN2W), FP(ENC1_BN2B), FP(ENC1_C3),
              A, B, R, WT, X1, PART, STATS, CNT + 1);

    k_maxpool<<<(int)(((long)32 * S2 + 255) / 256), 256, 0, stream>>>(X1, M1, M2, B, 32, 256, 256);
    res_block(stream, B, M2, 32, 64, 256, 256,
              FP(ENC2_C1), FP(ENC2_BN1W), FP(ENC2_BN1B),
              FP(ENC2_C2), FP(ENC2_BN2W), FP(ENC2_BN2B), FP(ENC2_C3),
              A, B, R, WT, X2, PART, STATS, CNT + 2);

    // ---- decoder level 0 (256 -> 512, 64ch -> 32ch) ----
    k_upsample_add<<<(int)(((long)32 * S1 + 255) / 256), 256, 0, stream>>>(
        X2, FP(DEC0_UP), X1, CM0, B, 64, 32, 512, 512);
    res_block(stream, B, CM0, 32, 32, 512, 512,
              FP(DEC0_C1), FP(DEC0_BN1W), FP(DEC0_BN1B),
              FP(DEC0_C2), FP(DEC0_BN2W), FP(DEC0_BN2B), FP(DEC0_C3),
              A, B, R, WT, D0, PART, STATS, CNT + 3);
    // predict head 0 -> coarse logits
    k_pred_stats<<<dim3(BPC, 32), 256, 0, stream>>>(D0, FP(PRED0_W1), 32, S1, PART);
    k_stats<<<1, 64, 0, stream>>>(PART, 32, CNT + 3, 1e-5f, STATS);
    k_pred_out<<<(S1 + 255) / 256, 256, 0, stream>>>(
        D0, FP(PRED0_W1), FP(PRED0_W2), STATS, CM0, logitsC, 32, 32, S1);

    // ---- decoder level 1 (512 -> 1024, 32ch -> 16ch) ----
    k_upsample_add<<<(int)(((long)16 * S0 + 255) / 256), 256, 0, stream>>>(
        D0, FP(DEC1_UP), X0, CM1, B, 32, 16, 1024, 1024);
    res_block(stream, B, CM1, 16, 16, 1024, 1024,
              FP(DEC1_C1), FP(DEC1_BN1W), FP(DEC1_BN1B),
              FP(DEC1_C2), FP(DEC1_BN2W), FP(DEC1_BN2B), FP(DEC1_C3),
              A, B, R, WT, xd_out, PART, STATS, CNT + 4);
    // predict head 1 -> fine logits
    k_pred_stats<<<dim3(BPC, 32), 256, 0, stream>>>(xd_out, FP(PRED1_W1), 16, S0, PART);
    k_stats<<<1, 64, 0, stream>>>(PART, 32, CNT + 4, 1e-5f, STATS);
    k_pred_out<<<(S0 + 255) / 256, 256, 0, stream>>>(
        xd_out, FP(PRED1_W1), FP(PRED1_W2), STATS, CM1, logitsF, 16, 32, S0);
#undef FP
}